// SNNetwork_35905926595236
// MI455X (gfx1250) — compile-verified
//
#include <hip/hip_runtime.h>
#include <hip/hip_bf16.h>
#include <stdint.h>

#define N_INPUT   500
#define N_HIDDEN  1500
#define N_OUTPUT  10
#define N_NEURONS 2010
#define N_LEARN   1510
#define ALPHA     4
#define MEMORY    10
#define K_DIM     (N_NEURONS * ALPHA)   // 8040
#define M_DIM     (N_LEARN * ALPHA)     // 6040

// Per-wave slot count (8 waves x 252 = 2016 >= 2010) and padded mask stride.
#define CHUNK       252
#define MASK_STRIDE 2016

typedef __attribute__((ext_vector_type(2))) float v2f;
typedef __attribute__((ext_vector_type(8))) float v8f;
typedef __attribute__((ext_vector_type(4))) int   v4i;

typedef __attribute__((address_space(1))) v4i gas_v4i;  // global (__device__)
typedef __attribute__((address_space(3))) v4i las_v4i;  // LDS (__local__)

#if defined(__gfx1250__) && \
    __has_builtin(__builtin_amdgcn_global_load_async_to_lds_b128) && \
    __has_builtin(__builtin_amdgcn_s_wait_asynccnt)
#define USE_ASYNC_LDS 1
#else
#define USE_ASYNC_LDS 0
#endif

// ---------------------------------------------------------------------------
// Kernel 1: exponential-filter traces (tiny).
// ---------------------------------------------------------------------------
__global__ void snn_trace_kernel(const float* __restrict__ hist,
                                 float* __restrict__ ff_trace,
                                 float* __restrict__ fb_trace) {
    int idx = blockIdx.x * blockDim.x + threadIdx.x;
    if (idx >= K_DIM) return;
    int n = idx >> 2;
    int a = idx & 3;
    const float* h = hist + (size_t)(n * ALPHA + a) * MEMORY;
    float ff = 0.0f, fb = 0.0f;
#pragma unroll
    for (int s = 0; s < MEMORY; ++s) {
        float hv = h[s];
        float dt = (float)(MEMORY - 1 - s);
        ff += hv * expf(-dt * 0.1f);   // tau_ff = 10
        fb += hv * expf(-dt * 0.2f);   // tau_fb = 5
    }
    ff_trace[idx] = ff;
    if (n >= N_INPUT) fb_trace[(n - N_INPUT) * ALPHA + a] = fb;
}

// ---------------------------------------------------------------------------
// Kernel 2: masked GEMV via V_WMMA_F32_16X16X4_F32.
// Block = 16 output rows (4 l's); 8 waves each run a fixed 252-slot K-slice
// (mask rows padded to stride 2016 with zeros so dead slots contribute 0).
// A fragment: global_load_b64 of W row pair at k = 4n + (lane<16?0:2).
// B fragment: ds_load_b64 broadcast of the trace pair at the same k.
// ---------------------------------------------------------------------------
__global__ void __launch_bounds__(256, 1)
snn_gemv_wmma_kernel(const float* __restrict__ W,
                     const int*   __restrict__ topo,
                     const float* __restrict__ ff_trace,
                     float*       __restrict__ ff_pot) {
    __shared__ float trace_lds[K_DIM];              // 32160 B
    __shared__ float mask_lds[4 * MASK_STRIDE];     // 32256 B (padded, pad = 0)
    __shared__ float red[8][16];

    const int tid    = threadIdx.x;
    const int m_base = blockIdx.x * 16;
    const int l_base = m_base >> 2;

    // ---- Stage trace into LDS (async DMA path when available) ----
#if USE_ASYNC_LDS
    for (int i = tid * 4; i < K_DIM; i += 256 * 4) {
        __builtin_amdgcn_global_load_async_to_lds_b128(
            (gas_v4i*)(ff_trace + i),
            (las_v4i*)(&trace_lds[i]),
            0, 0);
    }
#else
    for (int i = tid; i < K_DIM; i += 256) trace_lds[i] = ff_trace[i];
#endif

    // ---- Stage mask: topology -> float, diagonal + OOB + pad zeroing ----
    for (int i = tid; i < 4 * MASK_STRIDE; i += 256) {
        int lr = i / MASK_STRIDE;
        int n  = i - lr * MASK_STRIDE;
        int l  = l_base + lr;
        float mv = 0.0f;
        if (n < N_NEURONS && l < N_LEARN) {
            mv = (n == N_INPUT + l) ? 0.0f
                                    : (float)topo[(size_t)l * N_NEURONS + n];
        }
        mask_lds[i] = mv;
    }

#if USE_ASYNC_LDS
    __builtin_amdgcn_s_wait_asynccnt(0);
#endif
    __syncthreads();

    const int lane    = tid & 31;
    const int wave    = tid >> 5;
    const int row     = lane & 15;        // output row within tile
    const int hi      = lane >> 4;        // 0: K={0,1}, 1: K={2,3}
    const int l_local = row >> 2;
    int m      = m_base + row;
    int mclamp = (m < M_DIM) ? m : (M_DIM - 1);   // mask==0 kills OOB rows

    const v2f*   wrow = reinterpret_cast<const v2f*>(W + (size_t)mclamp * K_DIM);
    const v2f*   trow = reinterpret_cast<const v2f*>(trace_lds);
    const float* mrow = mask_lds + l_local * MASK_STRIDE;

    const int base = wave * CHUNK;        // fixed per-wave K slice

    v8f acc = {};
#pragma unroll 12
    for (int j = 0; j < CHUNK; ++j) {
        int s = base + j;                               // wave-uniform slot
        int n = (s < N_NEURONS) ? s : (N_NEURONS - 1);  // clamp addr; mask=0 in pad
        float mk = mrow[s];                             // padded row, pad = 0
        v2f a = wrow[2 * n + hi];                       // global_load_b64 (W stream)
        v2f t = trow[2 * n + hi];                       // ds_load_b64 (broadcast B)
        v2f av = a * mk;
        // D = A(16x4) x B(4x16, broadcast columns) + C
        acc = __builtin_amdgcn_wmma_f32_16x16x4_f32(
                  false, av, false, t, (short)0, acc, false, false);
    }

    // Broadcast-B => every column of D equals the GEMV result.
    // Lane 0 holds rows 0..7 in its 8 acc VGPRs, lane 16 holds rows 8..15.
    if (lane == 0) {
#pragma unroll
        for (int v = 0; v < 8; ++v) red[wave][v] = acc[v];
    } else if (lane == 16) {
#pragma unroll
        for (int v = 0; v < 8; ++v) red[wave][8 + v] = acc[v];
    }
    __syncthreads();

    if (tid < 16) {
        float s = 0.0f;
#pragma unroll
        for (int w = 0; w < 8; ++w) s += red[w][tid];
        int mo = m_base + tid;
        if (mo < M_DIM) ff_pot[mo] = s;
    }
}

// ---------------------------------------------------------------------------
// Kernel 3: finalize. fb term + bias, log-softmax over 5 logits, category
// selection (threefry2x32 Gumbel argmax for hidden; input_idx for outputs).
// ---------------------------------------------------------------------------
__device__ __forceinline__ uint32_t tf_rotl(uint32_t x, int r) {
    return (x << r) | (x >> (32 - r));
}

// JAX threefry2x32 with key = (0, 7)  (jax.random.key(7))
__device__ __forceinline__ void threefry2x32_key7(uint32_t x0, uint32_t x1,
                                                  uint32_t* o0, uint32_t* o1) {
    const uint32_t k0 = 0u, k1 = 7u;
    const uint32_t k2 = k0 ^ k1 ^ 0x1BD11BDAu;
    const int rA[4] = {13, 15, 26, 6};
    const int rB[4] = {17, 29, 16, 24};
    x0 += k0; x1 += k1;
#pragma unroll
    for (int i = 0; i < 4; ++i) { x0 += x1; x1 = tf_rotl(x1, rA[i]); x1 ^= x0; }
    x0 += k1; x1 += k2 + 1u;
#pragma unroll
    for (int i = 0; i < 4; ++i) { x0 += x1; x1 = tf_rotl(x1, rB[i]); x1 ^= x0; }
    x0 += k2; x1 += k0 + 2u;
#pragma unroll
    for (int i = 0; i < 4; ++i) { x0 += x1; x1 = tf_rotl(x1, rA[i]); x1 ^= x0; }
    x0 += k0; x1 += k1 + 3u;
#pragma unroll
    for (int i = 0; i < 4; ++i) { x0 += x1; x1 = tf_rotl(x1, rB[i]); x1 ^= x0; }
    x0 += k1; x1 += k2 + 4u;
#pragma unroll
    for (int i = 0; i < 4; ++i) { x0 += x1; x1 = tf_rotl(x1, rA[i]); x1 ^= x0; }
    x0 += k2; x1 += k0 + 5u;
    *o0 = x0; *o1 = x1;
}

__global__ void snn_final_kernel(const float* __restrict__ ff_pot,
                                 const float* __restrict__ fbw,
                                 const float* __restrict__ bias,
                                 const float* __restrict__ fb_trace,
                                 const int*   __restrict__ input_idx,
                                 float*       __restrict__ out) {
    int l = blockIdx.x * blockDim.x + threadIdx.x;
    if (l >= N_LEARN) return;

    float p[5];
    p[0] = 0.0f;
#pragma unroll
    for (int a = 0; a < 4; ++a) {
        int j = l * 4 + a;
        p[a + 1] = ff_pot[j] + fbw[j] * fb_trace[j] + bias[j];
    }

    float mx = p[0];
#pragma unroll
    for (int c = 1; c < 5; ++c) mx = fmaxf(mx, p[c]);
    float s = 0.0f;
#pragma unroll
    for (int c = 0; c < 5; ++c) s += expf(p[c] - mx);
    float lse = logf(s) + mx;

    int cat;
    if (l < N_HIDDEN) {
        // gumbel bits laid out row-major over (1500, 5); total 7500 words.
        // word j<3750 -> out0 of block j ; j>=3750 -> out1 of block j-3750.
        float best = -INFINITY;
        cat = 0;
#pragma unroll
        for (int c = 0; c < 5; ++c) {
            uint32_t j   = (uint32_t)(l * 5 + c);
            uint32_t blk = (j < 3750u) ? j : (j - 3750u);
            uint32_t o0, o1;
            threefry2x32_key7(blk, blk + 3750u, &o0, &o1);
            uint32_t bits = (j < 3750u) ? o0 : o1;
            float u = __uint_as_float((bits >> 9) | 0x3f800000u) - 1.0f;
            const float tiny = 1.17549435e-38f;
            u = fmaxf(u * (1.0f - tiny) + tiny, tiny);
            float g = -logf(-logf(u));
            float val = p[c] + g;
            if (val > best) { best = val; cat = c; }
        }
    } else {
        cat = input_idx[N_INPUT + (l - N_HIDDEN)];
        cat = cat < 0 ? 0 : (cat > 4 ? 4 : cat);
    }
    out[l] = p[cat] - lse;
}

// ---------------------------------------------------------------------------
// Launch
// ---------------------------------------------------------------------------
extern "C" void kernel_launch(void* const* d_in, const int* in_sizes, int n_in,
                              void* d_out, int out_size, void* d_ws, size_t ws_size,
                              hipStream_t stream) {
    const float* ffw  = (const float*)d_in[0];  // (1510,4,2010,4)
    const float* fbw  = (const float*)d_in[1];  // (1510,4)
    const float* bias = (const float*)d_in[2];  // (1510,4)
    const float* hist = (const float*)d_in[3];  // (2010,4,10)
    const int*   topo = (const int*)d_in[4];    // (1510,2010)
    const int*   iidx = (const int*)d_in[5];    // (510,)
    float* out = (float*)d_out;                 // (1510,)

    float* ws       = (float*)d_ws;
    float* ff_trace = ws;                       // 8040 floats
    float* fb_trace = ws + K_DIM;               // 6040 floats
    float* ff_pot   = ws + K_DIM + M_DIM;       // 6040 floats

    snn_trace_kernel<<<(K_DIM + 255) / 256, 256, 0, stream>>>(hist, ff_trace, fb_trace);
    snn_gemv_wmma_kernel<<<(M_DIM + 15) / 16, 256, 0, stream>>>(ffw, topo, ff_trace, ff_pot);
    snn_final_kernel<<<(N_LEARN + 255) / 256, 256, 0, stream>>>(ff_pot, fbw, bias,
                                                                fb_trace, iidx, out);
}